// IntraMA_49100066128304
// MI455X (gfx1250) — compile-verified
//
#include <hip/hip_runtime.h>
#include <math.h>

#define E_    4096
#define N_    4096
#define DIN   256
#define DEMB  64
#define H_    8
#define HD_   512
#define NET   4
#define NNT   2
#define SCALE_INV 0.125f   // 1/sqrt(64)
#define EPS_  1e-5f

typedef float v2f __attribute__((ext_vector_type(2)));
typedef float v8f __attribute__((ext_vector_type(8)));

static __device__ __forceinline__ v8f wmma4(v2f a, v2f b, v8f c) {
  // D = A(16x4 f32) * B(4x16 f32) + C(16x16 f32)
  return __builtin_amdgcn_wmma_f32_16x16x4_f32(false, a, false, b, (short)0, c, false, false);
}

#define V8F_ZERO ((v8f){0.f,0.f,0.f,0.f,0.f,0.f,0.f,0.f})

// ---------------------------------------------------------------------------
// LayerNorm: one wave per row of 256 floats (rows 0..E-1 = hyperedges, then nodes)
// ---------------------------------------------------------------------------
__global__ void ln_kernel(const float* __restrict__ hef, const float* __restrict__ nff,
                          const float* __restrict__ ge, const float* __restrict__ be,
                          const float* __restrict__ gn, const float* __restrict__ bn,
                          float* __restrict__ he_ln, float* __restrict__ nf_ln) {
  int wid  = (blockIdx.x * blockDim.x + threadIdx.x) >> 5;
  int lane = threadIdx.x & 31;
  const float *x, *g, *b; float* y;
  if (wid < E_) {
    x = hef + (size_t)wid * DIN; y = he_ln + (size_t)wid * DIN; g = ge; b = be;
  } else {
    int r = wid - E_;
    x = nff + (size_t)r * DIN; y = nf_ln + (size_t)r * DIN; g = gn; b = bn;
  }
  int col = lane * 8;
  float4 a0 = *(const float4*)(x + col);
  float4 a1 = *(const float4*)(x + col + 4);
  float s = a0.x + a0.y + a0.z + a0.w + a1.x + a1.y + a1.z + a1.w;
  float q = a0.x*a0.x + a0.y*a0.y + a0.z*a0.z + a0.w*a0.w
          + a1.x*a1.x + a1.y*a1.y + a1.z*a1.z + a1.w*a1.w;
  #pragma unroll
  for (int off = 1; off < 32; off <<= 1) {
    s += __shfl_xor(s, off, 32);
    q += __shfl_xor(q, off, 32);
  }
  float mean = s * (1.0f / DIN);
  float var  = q * (1.0f / DIN) - mean * mean;
  float rstd = rsqrtf(var + EPS_);
  float4 g0 = *(const float4*)(g + col);
  float4 g1 = *(const float4*)(g + col + 4);
  float4 b0 = *(const float4*)(b + col);
  float4 b1 = *(const float4*)(b + col + 4);
  float4 o0, o1;
  o0.x = (a0.x - mean) * rstd * g0.x + b0.x;
  o0.y = (a0.y - mean) * rstd * g0.y + b0.y;
  o0.z = (a0.z - mean) * rstd * g0.z + b0.z;
  o0.w = (a0.w - mean) * rstd * g0.w + b0.w;
  o1.x = (a1.x - mean) * rstd * g1.x + b1.x;
  o1.y = (a1.y - mean) * rstd * g1.y + b1.y;
  o1.z = (a1.z - mean) * rstd * g1.z + b1.z;
  o1.w = (a1.w - mean) * rstd * g1.w + b1.w;
  *(float4*)(y + col)     = o0;
  *(float4*)(y + col + 4) = o1;
}

// ---------------------------------------------------------------------------
// q projection: single K pass, 4 parallel accumulator chains (one per expert),
// per-row select by hyperedge type. No A-fragment array -> no spills.
// ---------------------------------------------------------------------------
__global__ void qproj_kernel(const float* __restrict__ he_ln, const int* __restrict__ etype,
                             const float* __restrict__ Wq, float* __restrict__ qout) {
  int wid  = (blockIdx.x * blockDim.x + threadIdx.x) >> 5;
  int lane = threadIdx.x & 31, lo = lane & 15, hi = lane >> 4;
  int et = wid >> 5, ct = wid & 31;        // 256 e-tiles x 32 c-tiles
  int eb = et * 16, cb = ct * 16;

  const float* arow = he_ln + (size_t)(eb + lo) * DIN + 2 * hi;
  const float* Bp0 = Wq + (size_t)0 * DIN * HD_ + cb + lo;
  const float* Bp1 = Wq + (size_t)1 * DIN * HD_ + cb + lo;
  const float* Bp2 = Wq + (size_t)2 * DIN * HD_ + cb + lo;
  const float* Bp3 = Wq + (size_t)3 * DIN * HD_ + cb + lo;

  v8f c0 = V8F_ZERO, c1 = V8F_ZERO, c2 = V8F_ZERO, c3 = V8F_ZERO;
  #pragma unroll 4
  for (int kb = 0; kb < 64; ++kb) {
    int kk = kb * 4 + 2 * hi;
    v2f af = *(const v2f*)(arow + kb * 4);
    v2f b0; b0.x = Bp0[(size_t)kk * HD_]; b0.y = Bp0[(size_t)(kk + 1) * HD_];
    v2f b1; b1.x = Bp1[(size_t)kk * HD_]; b1.y = Bp1[(size_t)(kk + 1) * HD_];
    v2f b2; b2.x = Bp2[(size_t)kk * HD_]; b2.y = Bp2[(size_t)(kk + 1) * HD_];
    v2f b3; b3.x = Bp3[(size_t)kk * HD_]; b3.y = Bp3[(size_t)(kk + 1) * HD_];
    c0 = wmma4(af, b0, c0);
    c1 = wmma4(af, b1, c1);
    c2 = wmma4(af, b2, c2);
    c3 = wmma4(af, b3, c3);
  }

  int tr[8];
  #pragma unroll
  for (int r = 0; r < 8; ++r) tr[r] = etype[eb + r + 8 * hi];

  #pragma unroll
  for (int r = 0; r < 8; ++r) {
    float acc = (tr[r] == 0) ? c0[r] : (tr[r] == 1) ? c1[r] : (tr[r] == 2) ? c2[r] : c3[r];
    qout[(size_t)(eb + r + 8 * hi) * HD_ + cb + lo] = acc;
  }
}

// ---------------------------------------------------------------------------
// k and v projections: 2 experts x 2 outputs = 4 parallel accumulator chains
// ---------------------------------------------------------------------------
__global__ void kvproj_kernel(const float* __restrict__ nf_ln, const int* __restrict__ ntype,
                              const float* __restrict__ Wk, const float* __restrict__ Wv,
                              float* __restrict__ kout, float* __restrict__ vout) {
  int wid  = (blockIdx.x * blockDim.x + threadIdx.x) >> 5;
  int lane = threadIdx.x & 31, lo = lane & 15, hi = lane >> 4;
  int nt = wid >> 5, ct = wid & 31;
  int nb = nt * 16, cb = ct * 16;

  const float* arow = nf_ln + (size_t)(nb + lo) * DIN + 2 * hi;
  const float* Bk0 = Wk + cb + lo;
  const float* Bk1 = Wk + (size_t)DIN * HD_ + cb + lo;
  const float* Bv0 = Wv + cb + lo;
  const float* Bv1 = Wv + (size_t)DIN * HD_ + cb + lo;

  v8f ck0 = V8F_ZERO, ck1 = V8F_ZERO, cv0 = V8F_ZERO, cv1 = V8F_ZERO;
  #pragma unroll 4
  for (int kb = 0; kb < 64; ++kb) {
    int kk = kb * 4 + 2 * hi;
    v2f af = *(const v2f*)(arow + kb * 4);
    v2f bk0; bk0.x = Bk0[(size_t)kk * HD_]; bk0.y = Bk0[(size_t)(kk + 1) * HD_];
    v2f bk1; bk1.x = Bk1[(size_t)kk * HD_]; bk1.y = Bk1[(size_t)(kk + 1) * HD_];
    v2f bv0; bv0.x = Bv0[(size_t)kk * HD_]; bv0.y = Bv0[(size_t)(kk + 1) * HD_];
    v2f bv1; bv1.x = Bv1[(size_t)kk * HD_]; bv1.y = Bv1[(size_t)(kk + 1) * HD_];
    ck0 = wmma4(af, bk0, ck0);
    ck1 = wmma4(af, bk1, ck1);
    cv0 = wmma4(af, bv0, cv0);
    cv1 = wmma4(af, bv1, cv1);
  }

  int tr[8];
  #pragma unroll
  for (int r = 0; r < 8; ++r) tr[r] = ntype[nb + r + 8 * hi];

  #pragma unroll
  for (int r = 0; r < 8; ++r) {
    size_t idx = (size_t)(nb + r + 8 * hi) * HD_ + cb + lo;
    kout[idx] = (tr[r] == 0) ? ck0[r] : ck1[r];
    vout[idx] = (tr[r] == 0) ? cv0[r] : cv1[r];
  }
}

// ---------------------------------------------------------------------------
// Attention: one wave per (head, 16-edge tile). Streams over N with online
// softmax; writes masked scores; accumulates O = softmax(S) @ V with WMMA;
// SiLU fused into epilogue. QK^T uses two interleaved accumulator chains.
// ---------------------------------------------------------------------------
__global__ void attn_kernel(const float* __restrict__ q, const float* __restrict__ k,
                            const float* __restrict__ v, const int* __restrict__ inc,
                            float* __restrict__ scores, float* __restrict__ attn_out) {
  __shared__ float pbuf[4][16 * 17];
  int wslot = (threadIdx.x >> 5) & 3;
  int wid  = (blockIdx.x * blockDim.x + threadIdx.x) >> 5;
  int lane = threadIdx.x & 31, lo = lane & 15, hi = lane >> 4;
  int h  = wid >> 8;          // 256 e-tiles per head
  int et = wid & 255;
  int eb = et * 16;

  const float* qrow = q + (size_t)(eb + lo) * HD_ + h * DEMB + 2 * hi;
  v2f qf[16];
  #pragma unroll
  for (int kb = 0; kb < 16; ++kb) qf[kb] = *(const v2f*)(qrow + kb * 4);

  v8f o[4];
  #pragma unroll
  for (int dc = 0; dc < 4; ++dc) o[dc] = V8F_ZERO;
  float m[8], lsum[8];
  #pragma unroll
  for (int r = 0; r < 8; ++r) { m[r] = -INFINITY; lsum[r] = 0.0f; }

  float* sc = scores + (size_t)h * E_ * N_;

  for (int nb = 0; nb < N_; nb += 16) {
    // S tile = Q (16x64) * K^T (64x16), two interleaved WMMA chains
    const float* krow = k + (size_t)(nb + lo) * HD_ + h * DEMB + 2 * hi;
    v8f ca = V8F_ZERO, cbv = V8F_ZERO;
    #pragma unroll
    for (int kb = 0; kb < 16; kb += 2) {
      v2f kf0 = *(const v2f*)(krow + kb * 4);
      v2f kf1 = *(const v2f*)(krow + kb * 4 + 4);
      ca  = wmma4(qf[kb],     kf0, ca);
      cbv = wmma4(qf[kb + 1], kf1, cbv);
    }

    // mask: inc[n, e] row-major [N, E]
    int mrow[8];
    size_t ibase = (size_t)(nb + lo) * E_ + eb + 8 * hi;
    #pragma unroll
    for (int r = 0; r < 8; ++r) mrow[r] = inc[ibase + r];

    float pv[8], al[8];
    #pragma unroll
    for (int r = 0; r < 8; ++r) {
      float s  = (ca[r] + cbv[r]) * SCALE_INV;
      float sm = (mrow[r] != 0) ? s : -INFINITY;
      // write masked pre-softmax score (output #2)
      sc[(size_t)(eb + r + 8 * hi) * N_ + nb + lo] = sm;
      // row max over the 16 lanes holding this row
      float tm = sm;
      tm = fmaxf(tm, __shfl_xor(tm, 1, 16));
      tm = fmaxf(tm, __shfl_xor(tm, 2, 16));
      tm = fmaxf(tm, __shfl_xor(tm, 4, 16));
      tm = fmaxf(tm, __shfl_xor(tm, 8, 16));
      float mn    = fmaxf(m[r], tm);
      float alpha = (m[r] > -INFINITY) ? __expf(m[r] - mn) : 0.0f;
      float p     = (sm   > -INFINITY) ? __expf(sm - mn)   : 0.0f;
      float ps = p;
      ps += __shfl_xor(ps, 1, 16);
      ps += __shfl_xor(ps, 2, 16);
      ps += __shfl_xor(ps, 4, 16);
      ps += __shfl_xor(ps, 8, 16);
      lsum[r] = lsum[r] * alpha + ps;
      m[r] = mn;
      pv[r] = p;
      al[r] = alpha;
    }

    // rescale O accumulator (same row<->VGPR mapping as C layout)
    #pragma unroll
    for (int dc = 0; dc < 4; ++dc)
      #pragma unroll
      for (int r = 0; r < 8; ++r) o[dc][r] *= al[r];

    // C-layout -> A-layout for P via padded LDS tile (same-wave DS is in-order)
    #pragma unroll
    for (int r = 0; r < 8; ++r) pbuf[wslot][(r + 8 * hi) * 17 + lo] = pv[r];
    asm volatile("" ::: "memory");
    v2f paf[4];
    #pragma unroll
    for (int kb2 = 0; kb2 < 4; ++kb2) {
      int kk = kb2 * 4 + 2 * hi;
      v2f t; t.x = pbuf[wslot][lo * 17 + kk]; t.y = pbuf[wslot][lo * 17 + kk + 1];
      paf[kb2] = t;
    }
    asm volatile("" ::: "memory");

    // O += P (16x16) * V (16x64): 4 parallel chains (one per d-chunk)
    const float* vbase = v + (size_t)nb * HD_ + h * DEMB;
    #pragma unroll
    for (int kb2 = 0; kb2 < 4; ++kb2) {
      int nl = kb2 * 4 + 2 * hi;
      #pragma unroll
      for (int dc = 0; dc < 4; ++dc) {
        v2f bf;
        bf.x = vbase[(size_t)nl * HD_ + dc * 16 + lo];
        bf.y = vbase[(size_t)(nl + 1) * HD_ + dc * 16 + lo];
        o[dc] = wmma4(paf[kb2], bf, o[dc]);
      }
    }
  }

  // normalize + SiLU fused, store [E, HD]
  #pragma unroll
  for (int r = 0; r < 8; ++r) {
    float inv = (lsum[r] > 0.0f) ? 1.0f / lsum[r] : 0.0f;
    size_t rowoff = (size_t)(eb + r + 8 * hi) * HD_ + h * DEMB;
    #pragma unroll
    for (int dc = 0; dc < 4; ++dc) {
      float val = o[dc][r] * inv;
      val = val / (1.0f + __expf(-val));   // silu
      attn_out[rowoff + dc * 16 + lo] = val;
    }
  }
}

// ---------------------------------------------------------------------------
// final = concat([he_ln, silu(out)]) @ Wo + bo   (K = 256 + 512 = 768)
// 4 parallel accumulator chains.
// ---------------------------------------------------------------------------
__global__ void final_kernel(const float* __restrict__ he_ln, const float* __restrict__ attn,
                             const float* __restrict__ Wo, const float* __restrict__ bo,
                             float* __restrict__ outp) {
  int wid  = (blockIdx.x * blockDim.x + threadIdx.x) >> 5;
  int lane = threadIdx.x & 31, lo = lane & 15, hi = lane >> 4;
  int et = wid >> 4, ct = wid & 15;       // 256 e-tiles x 16 c-tiles
  int eb = et * 16, cb = ct * 16;

  const float* ah = he_ln + (size_t)(eb + lo) * DIN + 2 * hi;
  const float* aa = attn  + (size_t)(eb + lo) * HD_ + 2 * hi;
  const float* Bp = Wo + cb + lo;

  v8f acc[4];
  #pragma unroll
  for (int i = 0; i < 4; ++i) acc[i] = V8F_ZERO;

  #pragma unroll 4
  for (int kb = 0; kb < 64; ++kb) {       // he part, k = 0..255
    int kk = kb * 4 + 2 * hi;
    v2f af = *(const v2f*)(ah + kb * 4);
    v2f bf; bf.x = Bp[(size_t)kk * DIN]; bf.y = Bp[(size_t)(kk + 1) * DIN];
    acc[kb & 3] = wmma4(af, bf, acc[kb & 3]);
  }
  #pragma unroll 4
  for (int kb = 64; kb < 192; ++kb) {     // silu(out) part, k = 256..767
    int kk = kb * 4 + 2 * hi;
    v2f af = *(const v2f*)(aa + (kb - 64) * 4);
    v2f bf; bf.x = Bp[(size_t)kk * DIN]; bf.y = Bp[(size_t)(kk + 1) * DIN];
    acc[kb & 3] = wmma4(af, bf, acc[kb & 3]);
  }
  float bias = bo[cb + lo];
  #pragma unroll
  for (int r = 0; r < 8; ++r) {
    float val = acc[0][r] + acc[1][r] + acc[2][r] + acc[3][r] + bias;
    outp[(size_t)(eb + r + 8 * hi) * DIN + cb + lo] = val;
  }
}

// ---------------------------------------------------------------------------
extern "C" void kernel_launch(void* const* d_in, const int* in_sizes, int n_in,
                              void* d_out, int out_size, void* d_ws, size_t ws_size,
                              hipStream_t stream) {
  const float* hef   = (const float*)d_in[0];
  const float* nff   = (const float*)d_in[1];
  const int*   inc   = (const int*)d_in[2];
  const int*   ntype = (const int*)d_in[3];
  const int*   etype = (const int*)d_in[4];
  const float* Wq    = (const float*)d_in[5];
  const float* Wk    = (const float*)d_in[6];
  const float* Wv    = (const float*)d_in[7];
  const float* Wo    = (const float*)d_in[8];
  const float* bo    = (const float*)d_in[9];
  const float* ge    = (const float*)d_in[10];
  const float* be    = (const float*)d_in[11];
  const float* gn    = (const float*)d_in[12];
  const float* bn    = (const float*)d_in[13];

  float* outp   = (float*)d_out;
  float* scores = outp + (size_t)E_ * DIN;      // output #2 region

  float* ws    = (float*)d_ws;
  float* he_ln = ws;
  float* nf_ln = he_ln + (size_t)E_ * DIN;
  float* qbuf  = nf_ln + (size_t)N_ * DIN;
  float* kbuf  = qbuf + (size_t)E_ * HD_;
  float* vbuf  = kbuf + (size_t)N_ * HD_;
  float* abuf  = vbuf + (size_t)N_ * HD_;       // silu(attention out), [E, HD]

  // LN: 8192 rows, 8 waves/block
  ln_kernel<<<(E_ + N_) / 8, 256, 0, stream>>>(hef, nff, ge, be, gn, bn, he_ln, nf_ln);
  // q: 256*32 tile-waves, 4 waves/block
  qproj_kernel<<<2048, 128, 0, stream>>>(he_ln, etype, Wq, qbuf);
  // k,v: 256*32 tile-waves
  kvproj_kernel<<<2048, 128, 0, stream>>>(nf_ln, ntype, Wk, Wv, kbuf, vbuf);
  // attention: 8 heads * 256 e-tiles = 2048 waves
  attn_kernel<<<512, 128, 0, stream>>>(qbuf, kbuf, vbuf, inc, scores, abuf);
  // final GEMM: 256*16 tile-waves
  final_kernel<<<1024, 128, 0, stream>>>(he_ln, abuf, Wo, bo, outp);
}